// ConformerEncoder_78374563218049
// MI455X (gfx1250) — compile-verified
//
#include <hip/hip_runtime.h>
#include <hip/hip_bf16.h>
#include <math.h>

typedef __attribute__((ext_vector_type(2))) float v2f;
typedef __attribute__((ext_vector_type(8))) float v8f;

// ---------------------------------------------------------------------------
// Generic direct conv (k x k, pad) fused with MFM (max over channel halves)
// and optional 2x2/2x2 max-pool (floor semantics). NCHW layout. One thread
// per (pooled) output element.
// ---------------------------------------------------------------------------
__global__ void conv_mfm_pool_kernel(const float* __restrict__ in,
                                     const float* __restrict__ wgt,
                                     const float* __restrict__ bias,
                                     float* __restrict__ out,
                                     int B, int Cin, int CoutHalf,
                                     int Hin, int Win, int K, int pad, int dopool) {
    int Hout = dopool ? (Hin >> 1) : Hin;
    int Wout = dopool ? (Win >> 1) : Win;
    int total = B * CoutHalf * Hout * Wout;
    int idx = blockIdx.x * blockDim.x + threadIdx.x;
    if (idx >= total) return;
    int wo = idx % Wout;
    int ho = (idx / Wout) % Hout;
    int o  = (idx / (Wout * Hout)) % CoutHalf;
    int b  = idx / (Wout * Hout * CoutHalf);

    int span = dopool ? 2 : 1;
    float best = -3.4e38f;
    for (int dy = 0; dy < span; ++dy) {
        for (int dx = 0; dx < span; ++dx) {
            int h = dopool ? (2 * ho + dy) : ho;
            int w = dopool ? (2 * wo + dx) : wo;
            float acc1 = bias[o];
            float acc2 = bias[o + CoutHalf];
            for (int ci = 0; ci < Cin; ++ci) {
                const float* inp = in + ((size_t)(b * Cin + ci) * Hin) * Win;
                const float* w1p = wgt + ((size_t)(o * Cin + ci) * K) * K;
                const float* w2p = wgt + ((size_t)((o + CoutHalf) * Cin + ci) * K) * K;
                for (int ky = 0; ky < K; ++ky) {
                    int ih = h + ky - pad;
                    if (ih < 0 || ih >= Hin) continue;
                    for (int kx = 0; kx < K; ++kx) {
                        int iw = w + kx - pad;
                        if (iw < 0 || iw >= Win) continue;
                        float v = inp[ih * Win + iw];
                        acc1 = fmaf(v, w1p[ky * K + kx], acc1);
                        acc2 = fmaf(v, w2p[ky * K + kx], acc2);
                    }
                }
            }
            float m = fmaxf(acc1, acc2);
            best = fmaxf(best, m);
        }
    }
    out[idx] = best;
}

// ---------------------------------------------------------------------------
// 1x1 conv as f32 WMMA GEMM, fused MFM. A = activations (M pixels x Cin),
// column-per-channel in NCHW (A[m][k] = in[b][k][m]); B = weights
// (B[k][n] = w[n][k]). One wave per 16x16 output tile; two accumulators
// (out channel n and n+CoutHalf) so MFM max happens in the epilogue.
// Fragment layouts per CDNA5 ISA 7.12.2 (16x4 f32 A; mirrored B; 16x16 f32 D).
// Requires M % 16 == 0, Cin % 4 == 0 (all stages satisfy this).
// ---------------------------------------------------------------------------
__global__ void conv1x1_mfm_wmma_kernel(const float* __restrict__ in,
                                        const float* __restrict__ wgt,
                                        const float* __restrict__ bias,
                                        float* __restrict__ out,
                                        int M, int Cin, int CoutHalf) {
    int lane = threadIdx.x;           // 0..31 (wave32)
    int ml = lane & 15;               // A: M row / B,D: N column
    int kh = lane >> 4;               // K-half selector
    int m0 = blockIdx.x * 16;
    int n0 = blockIdx.y * 16;
    int b  = blockIdx.z;

    const float* inb = in + (size_t)b * Cin * M;
    int och1 = n0 + ml;
    int och2 = och1 + CoutHalf;

    v8f acc1 = {};
    v8f acc2 = {};
    for (int k0 = 0; k0 < Cin; k0 += 4) {
        int kk = k0 + 2 * kh;
        v2f a;
        a.x = inb[(size_t)kk * M + m0 + ml];
        a.y = inb[(size_t)(kk + 1) * M + m0 + ml];
        v2f b1;
        b1.x = wgt[och1 * Cin + kk];
        b1.y = wgt[och1 * Cin + kk + 1];
        v2f b2;
        b2.x = wgt[och2 * Cin + kk];
        b2.y = wgt[och2 * Cin + kk + 1];
        acc1 = __builtin_amdgcn_wmma_f32_16x16x4_f32(false, a, false, b1,
                                                     (short)0, acc1, false, false);
        acc2 = __builtin_amdgcn_wmma_f32_16x16x4_f32(false, a, false, b2,
                                                     (short)0, acc2, false, false);
    }
    float bi1 = bias[och1];
    float bi2 = bias[och2];
    float* outb = out + (size_t)b * CoutHalf * M;
    for (int r = 0; r < 8; ++r) {
        int row = m0 + r + 8 * kh;                   // D: M = vgpr + 8*(lane>=16)
        outb[(size_t)(n0 + ml) * M + row] = fmaxf(acc1[r] + bi1, acc2[r] + bi2);
    }
}

// ---------------------------------------------------------------------------
// BatchNorm (affine=False, train-mode stats over N,H,W)
// ---------------------------------------------------------------------------
__global__ void bn_stats_kernel(const float* __restrict__ in, float* __restrict__ meanb,
                                float* __restrict__ rstdb, int B, int C, int HW) {
    int c = blockIdx.x;
    __shared__ float ssum[256];
    __shared__ float ssq[256];
    float s = 0.f, q = 0.f;
    int N = B * HW;
    for (int i = threadIdx.x; i < N; i += blockDim.x) {
        int b = i / HW;
        int p = i - b * HW;
        float v = in[((size_t)(b * C + c)) * HW + p];
        s += v;
        q += v * v;
    }
    ssum[threadIdx.x] = s;
    ssq[threadIdx.x] = q;
    __syncthreads();
    for (int st = 128; st > 0; st >>= 1) {
        if (threadIdx.x < st) {
            ssum[threadIdx.x] += ssum[threadIdx.x + st];
            ssq[threadIdx.x] += ssq[threadIdx.x + st];
        }
        __syncthreads();
    }
    if (threadIdx.x == 0) {
        float m = ssum[0] / (float)N;
        float var = ssq[0] / (float)N - m * m;
        meanb[c] = m;
        rstdb[c] = rsqrtf(var + 1e-5f);
    }
}

__global__ void bn_apply_kernel(float* __restrict__ x, const float* __restrict__ meanb,
                                const float* __restrict__ rstdb, int C, int HW, int total) {
    int i = blockIdx.x * blockDim.x + threadIdx.x;
    if (i >= total) return;
    int c = (i / HW) % C;
    x[i] = (x[i] - meanb[c]) * rstdb[c];
}

// ---------------------------------------------------------------------------
// SE block: GAP -> Linear(C, C/2) -> ReLU -> Linear(C/2, C) -> sigmoid -> scale
// ---------------------------------------------------------------------------
__global__ void se_gap_kernel(const float* __restrict__ in, float* __restrict__ y, int HW) {
    int bc = blockIdx.x;   // b*C + c
    __shared__ float ssum[256];
    const float* p = in + (size_t)bc * HW;
    float s = 0.f;
    for (int i = threadIdx.x; i < HW; i += blockDim.x) s += p[i];
    ssum[threadIdx.x] = s;
    __syncthreads();
    for (int st = 128; st > 0; st >>= 1) {
        if (threadIdx.x < st) ssum[threadIdx.x] += ssum[threadIdx.x + st];
        __syncthreads();
    }
    if (threadIdx.x == 0) y[bc] = ssum[0] / (float)HW;
}

__global__ void se_mlp_kernel(const float* __restrict__ y, const float* __restrict__ w1,
                              const float* __restrict__ w2, float* __restrict__ sb, int C) {
    int b = blockIdx.x;
    int t = threadIdx.x;
    __shared__ float ysh[64];
    __shared__ float hsh[32];
    int Ch = C >> 1;
    if (t < C) ysh[t] = y[b * C + t];
    __syncthreads();
    if (t < Ch) {
        float a = 0.f;
        for (int c = 0; c < C; ++c) a = fmaf(w1[t * C + c], ysh[c], a);
        hsh[t] = fmaxf(a, 0.f);
    }
    __syncthreads();
    if (t < C) {
        float a = 0.f;
        for (int r = 0; r < Ch; ++r) a = fmaf(w2[t * Ch + r], hsh[r], a);
        sb[b * C + t] = 1.f / (1.f + expf(-a));
    }
}

__global__ void se_scale_kernel(float* __restrict__ x, const float* __restrict__ sb,
                                int HW, int total) {
    int i = blockIdx.x * blockDim.x + threadIdx.x;
    if (i >= total) return;
    x[i] *= sb[i / HW];
}

// ---------------------------------------------------------------------------
// (B,32,64,3) NCHW -> (B,64,96) tokens:  xatt[b][t][c*3+f] = s9[b][c][t][f]
// ---------------------------------------------------------------------------
__global__ void make_xatt_kernel(const float* __restrict__ s9, float* __restrict__ xatt) {
    int i = blockIdx.x * blockDim.x + threadIdx.x;
    const int total = 16 * 64 * 96;
    if (i >= total) return;
    int d = i % 96;
    int t = (i / 96) % 64;
    int b = i / (96 * 64);
    int c = d / 3;
    int f = d - c * 3;
    xatt[i] = s9[(((size_t)(b * 32 + c)) * 64 + t) * 3 + f];
}

// ---------------------------------------------------------------------------
// Q/K/V projections as f32 WMMA GEMM: (1024 x 96) @ W^T (96 x 96) + bias.
// grid.z selects which projection; q gets the 1/sqrt(d) scale folded in.
// ---------------------------------------------------------------------------
__global__ void qkv_wmma_kernel(const float* __restrict__ x,
                                const float* __restrict__ wq, const float* __restrict__ wk,
                                const float* __restrict__ wv,
                                const float* __restrict__ bq, const float* __restrict__ bk,
                                const float* __restrict__ bv,
                                float* __restrict__ qo, float* __restrict__ ko,
                                float* __restrict__ vo) {
    const float* W;
    const float* bias;
    float* out;
    float scale;
    if (blockIdx.z == 0) { W = wq; bias = bq; out = qo; scale = 0.20412414523193154f; }
    else if (blockIdx.z == 1) { W = wk; bias = bk; out = ko; scale = 1.f; }
    else { W = wv; bias = bv; out = vo; scale = 1.f; }

    int lane = threadIdx.x;
    int ml = lane & 15;
    int kh = lane >> 4;
    int m0 = blockIdx.x * 16;
    int n0 = blockIdx.y * 16;

    v8f acc = {};
    for (int k0 = 0; k0 < 96; k0 += 4) {
        int kk = k0 + 2 * kh;
        v2f a;
        a.x = x[(m0 + ml) * 96 + kk];
        a.y = x[(m0 + ml) * 96 + kk + 1];
        v2f bf;
        bf.x = W[(n0 + ml) * 96 + kk];
        bf.y = W[(n0 + ml) * 96 + kk + 1];
        acc = __builtin_amdgcn_wmma_f32_16x16x4_f32(false, a, false, bf,
                                                    (short)0, acc, false, false);
    }
    for (int r = 0; r < 8; ++r) {
        int row = m0 + r + 8 * kh;
        int col = n0 + ml;
        out[row * 96 + col] = (acc[r] + bias[col]) * scale;
    }
}

// ---------------------------------------------------------------------------
// Banded (|i-j|<=2) softmax attention + residual. One thread per (b,h,q).
// ---------------------------------------------------------------------------
__global__ void local_attn_kernel(const float* __restrict__ q, const float* __restrict__ k,
                                  const float* __restrict__ v, const float* __restrict__ xatt,
                                  float* __restrict__ out) {
    int i = blockIdx.x * blockDim.x + threadIdx.x;   // 16*4*64 = 4096
    if (i >= 4096) return;
    int qp = i % 64;
    int h = (i / 64) % 4;
    int b = i / (64 * 4);
    int base = (b * 64 + qp) * 96 + h * 24;
    int j0 = qp - 2 < 0 ? 0 : qp - 2;
    int j1 = qp + 2 > 63 ? 63 : qp + 2;
    float sc[5];
    float mx = -3.4e38f;
    for (int j = j0; j <= j1; ++j) {
        const float* kj = k + (b * 64 + j) * 96 + h * 24;
        float d = 0.f;
        for (int e = 0; e < 24; ++e) d = fmaf(q[base + e], kj[e], d);
        sc[j - j0] = d;
        mx = fmaxf(mx, d);
    }
    float sum = 0.f;
    for (int j = j0; j <= j1; ++j) {
        float e = expf(sc[j - j0] - mx);
        sc[j - j0] = e;
        sum += e;
    }
    float inv = 1.f / sum;
    for (int e = 0; e < 24; ++e) {
        float acc = 0.f;
        for (int j = j0; j <= j1; ++j)
            acc = fmaf(sc[j - j0], v[(b * 64 + j) * 96 + h * 24 + e], acc);
        out[base + e] = acc * inv + xatt[base + e];
    }
}

// ---------------------------------------------------------------------------
// Host-side sequencing
// ---------------------------------------------------------------------------
static void run_conv(const float* in, const float* w, const float* bias, float* out,
                     int Cin, int CoutHalf, int Hin, int Win, int K, int pad, int dopool,
                     hipStream_t s) {
    int Hout = dopool ? Hin / 2 : Hin;
    int Wout = dopool ? Win / 2 : Win;
    int total = 16 * CoutHalf * Hout * Wout;
    conv_mfm_pool_kernel<<<(total + 255) / 256, 256, 0, s>>>(
        in, w, bias, out, 16, Cin, CoutHalf, Hin, Win, K, pad, dopool);
}

static void run_conv1x1(const float* in, const float* w, const float* bias, float* out,
                        int M, int Cin, int CoutHalf, hipStream_t s) {
    dim3 grid(M / 16, CoutHalf / 16, 16);
    conv1x1_mfm_wmma_kernel<<<grid, 32, 0, s>>>(in, w, bias, out, M, Cin, CoutHalf);
}

static void run_bn(float* buf, int C, int HW, float* meanb, float* rstdb, hipStream_t s) {
    int total = 16 * C * HW;
    bn_stats_kernel<<<C, 256, 0, s>>>(buf, meanb, rstdb, 16, C, HW);
    bn_apply_kernel<<<(total + 255) / 256, 256, 0, s>>>(buf, meanb, rstdb, C, HW, total);
}

static void run_se(float* buf, int C, int HW, const float* w1, const float* w2,
                   float* ybuf, float* sbuf, hipStream_t s) {
    se_gap_kernel<<<16 * C, 256, 0, s>>>(buf, ybuf, HW);
    se_mlp_kernel<<<16, 64, 0, s>>>(ybuf, w1, w2, sbuf, C);
    int total = 16 * C * HW;
    se_scale_kernel<<<(total + 255) / 256, 256, 0, s>>>(buf, sbuf, HW, total);
}

extern "C" void kernel_launch(void* const* d_in, const int* in_sizes, int n_in,
                              void* d_out, int out_size, void* d_ws, size_t ws_size,
                              hipStream_t stream) {
    // Input indices per setup_inputs() dict order: x, then c1w,c1b..c9w,c9b,
    // se1a,se1b..se8a,se8b, wq,wk,wv, bq,bk,bv.
    const float* x = (const float*)d_in[0];
#define CW(i) ((const float*)d_in[1 + 2 * ((i) - 1)])
#define CB(i) ((const float*)d_in[2 + 2 * ((i) - 1)])
#define SEA(i) ((const float*)d_in[19 + 2 * ((i) - 1)])
#define SEB(i) ((const float*)d_in[20 + 2 * ((i) - 1)])
    const float* wq = (const float*)d_in[35];
    const float* wk = (const float*)d_in[36];
    const float* wv = (const float*)d_in[37];
    const float* bq = (const float*)d_in[38];
    const float* bk = (const float*)d_in[39];
    const float* bv = (const float*)d_in[40];

    // Workspace carve-up (floats). Peak buffer is (16,32,512,30) = 7,864,320.
    float* bufA = (float*)d_ws;
    float* bufB = bufA + 7864320;
    float* xatt = bufB + 7864320;
    float* qb = xatt + 98304;
    float* kb = qb + 98304;
    float* vb = kb + 98304;
    float* ybuf = vb + 98304;       // 1024
    float* sbuf = ybuf + 1024;      // 1024
    float* meanb = sbuf + 1024;     // 128
    float* rstdb = meanb + 128;     // 128

    // Stage 1: conv(1->64,k5,p2)+MFM+pool -> (16,32,512,30); SE(32)
    run_conv(x, CW(1), CB(1), bufA, 1, 32, 1024, 60, 5, 2, 1, stream);
    run_se(bufA, 32, 512 * 30, SEA(1), SEB(1), ybuf, sbuf, stream);

    // Stage 2: 1x1 (32->64)+MFM [WMMA] -> (16,32,512,30); BN; SE(32)
    run_conv1x1(bufA, CW(2), CB(2), bufB, 512 * 30, 32, 32, stream);
    run_bn(bufB, 32, 512 * 30, meanb, rstdb, stream);
    run_se(bufB, 32, 512 * 30, SEA(2), SEB(2), ybuf, sbuf, stream);

    // Stage 3: conv(32->96,k3,p1)+MFM+pool -> (16,48,256,15); BN; SE(48)
    run_conv(bufB, CW(3), CB(3), bufA, 32, 48, 512, 30, 3, 1, 1, stream);
    run_bn(bufA, 48, 256 * 15, meanb, rstdb, stream);
    run_se(bufA, 48, 256 * 15, SEA(3), SEB(3), ybuf, sbuf, stream);

    // Stage 4: 1x1 (48->96)+MFM [WMMA] -> (16,48,256,15); BN; SE(48)
    run_conv1x1(bufA, CW(4), CB(4), bufB, 256 * 15, 48, 48, stream);
    run_bn(bufB, 48, 256 * 15, meanb, rstdb, stream);
    run_se(bufB, 48, 256 * 15, SEA(4), SEB(4), ybuf, sbuf, stream);

    // Stage 5: conv(48->128,k3,p1)+MFM+pool -> (16,64,128,7); SE(64)
    run_conv(bufB, CW(5), CB(5), bufA, 48, 64, 256, 15, 3, 1, 1, stream);
    run_se(bufA, 64, 128 * 7, SEA(5), SEB(5), ybuf, sbuf, stream);

    // Stage 6: 1x1 (64->128)+MFM [WMMA] -> (16,64,128,7); BN; SE(64)
    run_conv1x1(bufA, CW(6), CB(6), bufB, 128 * 7, 64, 64, stream);
    run_bn(bufB, 64, 128 * 7, meanb, rstdb, stream);
    run_se(bufB, 64, 128 * 7, SEA(6), SEB(6), ybuf, sbuf, stream);

    // Stage 7: conv(64->64,k3,p1)+MFM -> (16,32,128,7); BN; SE(32)
    run_conv(bufB, CW(7), CB(7), bufA, 64, 32, 128, 7, 3, 1, 0, stream);
    run_bn(bufA, 32, 128 * 7, meanb, rstdb, stream);
    run_se(bufA, 32, 128 * 7, SEA(7), SEB(7), ybuf, sbuf, stream);

    // Stage 8: 1x1 (32->64)+MFM [WMMA] -> (16,32,128,7); BN; SE(32)
    run_conv1x1(bufA, CW(8), CB(8), bufB, 128 * 7, 32, 32, stream);
    run_bn(bufB, 32, 128 * 7, meanb, rstdb, stream);
    run_se(bufB, 32, 128 * 7, SEA(8), SEB(8), ybuf, sbuf, stream);

    // Stage 9: conv(32->64,k3,p1)+MFM+pool -> (16,32,64,3)
    run_conv(bufB, CW(9), CB(9), bufA, 32, 32, 128, 7, 3, 1, 1, stream);

    // Tokens + attention
    make_xatt_kernel<<<(16 * 64 * 96 + 255) / 256, 256, 0, stream>>>(bufA, xatt);
    {
        dim3 grid(1024 / 16, 96 / 16, 3);
        qkv_wmma_kernel<<<grid, 32, 0, stream>>>(xatt, wq, wk, wv, bq, bk, bv, qb, kb, vb);
    }
    local_attn_kernel<<<(4096 + 255) / 256, 256, 0, stream>>>(qb, kb, vb, xatt, (float*)d_out);
#undef CW
#undef CB
#undef SEA
#undef SEB
}